// GNNDecoder_71717363908961
// MI455X (gfx1250) — compile-verified
//
#include <hip/hip_runtime.h>

typedef float v2f __attribute__((ext_vector_type(2)));
typedef float v8f __attribute__((ext_vector_type(8)));

#define LN_EPS 1e-5f

// ---------------------------------------------------------------- zero fill
__global__ void zero_kernel(float* __restrict__ p, int n) {
    int i = blockIdx.x * blockDim.x + threadIdx.x;
    if (i < n) p[i] = 0.0f;
}

// ------------------------------------------------- pad cols 5 -> 8 (zeros)
__global__ void pad5to8_kernel(const float* __restrict__ in, float* __restrict__ out, int rows) {
    int i = blockIdx.x * blockDim.x + threadIdx.x;
    if (i < rows * 8) {
        const int r = i >> 3;
        const int c = i & 7;
        out[i] = (c < 5) ? in[r * 5 + c] : 0.0f;
    }
}

// -------------------------------------------------- edge scatter-add (agg)
// feature width = 1<<kshift; 256-thread blocks cover (256>>kshift) edges
__global__ void scatter_add_kernel(const float* __restrict__ h,
                                   const int* __restrict__ src,
                                   const int* __restrict__ dst,
                                   float* __restrict__ agg,
                                   int E, int kshift) {
    const int din = 1 << kshift;
    const int epb = 256 >> kshift;
    const int e = blockIdx.x * epb + (threadIdx.x >> kshift);
    if (e >= E) return;
    const int k = threadIdx.x & (din - 1);
    const float v = h[((size_t)src[e] << kshift) + k];
    atomicAdd(&agg[((size_t)dst[e] << kshift) + k], v);
}

// -------------------------------------------------------- fused GCN layer
// out = ReLU(LN(agg @ Wrel^T + h @ Wroot^T + brel))
// One block = 16 rows x DOUT cols.  NWAVES = min(4, DOUT/16) wave32 waves.
// Wave w owns column tiles ct = w + NWAVES*t, t in [0, TPW) -- exact cover,
// so the K-loop is branch-free: global_load_b64 pairs + v_wmma only.
template <int DIN, int DOUT>
__global__ void gcn_gemm_ln_relu(const float* __restrict__ h,
                                 const float* __restrict__ agg,
                                 const float* __restrict__ Wrel,
                                 const float* __restrict__ Wroot,
                                 const float* __restrict__ brel,
                                 const float* __restrict__ gamma,
                                 const float* __restrict__ beta,
                                 float* __restrict__ out,
                                 int N) {
    constexpr int NWAVES = (DOUT / 16 < 4) ? (DOUT / 16) : 4;
    constexpr int NT = NWAVES * 32;
    constexpr int TPW = DOUT / (16 * NWAVES);
    constexpr int PPR = NT / 16;  // partial reducers per row

    __shared__ float smem[16 * DOUT + 2 * NT + 32];

    const int tid = threadIdx.x;
    __builtin_assume(tid < NT);
    const int wave = tid >> 5;
    const int lane = tid & 31;
    const int half = lane >> 4;   // A/B f32 layout: K pair select
    const int l16  = lane & 15;
    const int rowBase = blockIdx.x * 16;
    int aRow = rowBase + l16;
    if (aRow > N - 1) aRow = N - 1;          // branchless clamp; stores guarded

    const size_t aOff = (size_t)aRow * DIN + half * 2;

    v8f acc[TPW] = {};

    #pragma unroll
    for (int phase = 0; phase < 2; ++phase) {
        const float* __restrict__ abase = (phase ? h : agg) + aOff;
        const float* __restrict__ W     = (phase ? Wroot : Wrel);
        #pragma unroll 8
        for (int k0 = 0; k0 < DIN; k0 += 4) {
            // A fragment: lane holds A[M=l16][K=k0+2*half .. +1]  (one b64 load)
            const v2f A = *(const v2f*)(abase + k0);
            #pragma unroll
            for (int t = 0; t < TPW; ++t) {
                const int o = (wave + NWAVES * t) * 16 + l16;
                // B fragment: B[k][n] = W[n][k]  (one b64 load)
                const v2f B = *(const v2f*)(W + (size_t)o * DIN + half * 2 + k0);
                acc[t] = __builtin_amdgcn_wmma_f32_16x16x4_f32(
                    false, A, false, B, (short)0, acc[t], false, false);
            }
        }
    }

    // ---- epilogue: bias into LDS tile, LayerNorm, ReLU, store
    float* tile = smem;                  // 16 * DOUT
    float* redS = smem + 16 * DOUT;      // NT
    float* redQ = redS + NT;             // NT
    float* muA  = redQ + NT;             // 16
    float* rsA  = muA + 16;              // 16

    #pragma unroll
    for (int t = 0; t < TPW; ++t) {
        const int o = (wave + NWAVES * t) * 16 + l16;
        const float bo = brel[o];
        #pragma unroll
        for (int i = 0; i < 8; ++i) {
            const int m = i + 8 * half;  // C layout: VGPR i, lane half -> M
            tile[m * DOUT + o] = acc[t][i] + bo;
        }
    }
    __syncthreads();

    {   // PPR partial reducers per row
        const int r = tid & 15;
        const int p = tid >> 4;
        float s = 0.0f, q = 0.0f;
        #pragma unroll 4
        for (int o = p; o < DOUT; o += PPR) {
            const float v = tile[r * DOUT + o];
            s += v; q += v * v;
        }
        redS[r * PPR + p] = s;
        redQ[r * PPR + p] = q;
    }
    __syncthreads();
    if (tid < 16) {
        float s = 0.0f, q = 0.0f;
        #pragma unroll
        for (int p = 0; p < PPR; ++p) { s += redS[tid * PPR + p]; q += redQ[tid * PPR + p]; }
        constexpr float inv = 1.0f / (float)DOUT;
        const float mu  = s * inv;
        const float var = q * inv - mu * mu;
        muA[tid] = mu;
        rsA[tid] = rsqrtf(var + LN_EPS);
    }
    __syncthreads();

    #pragma unroll 4
    for (int e = tid; e < 16 * DOUT; e += NT) {
        const int r = e / DOUT;          // DOUT is a power of two -> shift
        const int o = e - r * DOUT;
        if (rowBase + r < N) {
            const float v = (tile[e] - muA[r]) * rsA[r] * gamma[o] + beta[o];
            out[(size_t)(rowBase + r) * DOUT + o] = fmaxf(v, 0.0f);
        }
    }
}

template <int DIN, int DOUT>
static void launch_gcn(const float* h, const float* agg, const float* Wrel,
                       const float* Wroot, const float* brel, const float* g,
                       const float* b, float* out, int N, hipStream_t s) {
    constexpr int NWAVES = (DOUT / 16 < 4) ? (DOUT / 16) : 4;
    gcn_gemm_ln_relu<DIN, DOUT><<<(N + 15) / 16, NWAVES * 32, 0, s>>>(
        h, agg, Wrel, Wroot, brel, g, b, out, N);
}

// --------------------------------------------------------- global mean pool
__global__ void pool_accum_kernel(const float* __restrict__ h,
                                  const int* __restrict__ batch,
                                  float* __restrict__ pool,
                                  float* __restrict__ cnt) {
    const int n = blockIdx.x;
    const int k = threadIdx.x;   // 128
    const int g = batch[n];
    atomicAdd(&pool[(size_t)g * 128 + k], h[(size_t)n * 128 + k]);
    if (k == 0) atomicAdd(&cnt[g], 1.0f);
}

__global__ void pool_div_kernel(float* __restrict__ pool,
                                const float* __restrict__ cnt, int n) {
    const int i = blockIdx.x * blockDim.x + threadIdx.x;
    if (i < n) pool[i] /= fmaxf(cnt[i >> 7], 1.0f);
}

// ----------------------------------------------------------------- tiny MLP
__global__ void mlp_kernel(const float* __restrict__ in,
                           const float* __restrict__ W,
                           const float* __restrict__ b,
                           float* __restrict__ out,
                           int din, int dout, int doRelu) {
    const int g = blockIdx.x;
    const int o = threadIdx.x;
    float s = b[o];
    const float* row = in + (size_t)g * din;
    const float* w   = W  + (size_t)o * din;
    for (int k = 0; k < din; ++k) s += row[k] * w[k];
    if (doRelu) s = fmaxf(s, 0.0f);
    out[(size_t)g * dout + o] = s;
}

// ---------------------------------------------------------------- launcher
extern "C" void kernel_launch(void* const* d_in, const int* in_sizes, int n_in,
                              void* d_out, int out_size, void* d_ws, size_t ws_size,
                              hipStream_t stream) {
    const float* x     = (const float*)d_in[0];
    const int*   ei    = (const int*)d_in[1];
    const int*   batch = (const int*)d_in[2];
    const int N = in_sizes[2];
    const int E = in_sizes[1] / 2;
    const int G = 256;
    const int* src = ei;
    const int* dst = ei + E;

    const float *Wrel[5], *brel[5], *Wroot[5], *gam[5], *bet[5];
    for (int l = 0; l < 5; ++l) {
        Wrel[l]  = (const float*)d_in[3 + 3 * l];
        brel[l]  = (const float*)d_in[4 + 3 * l];
        Wroot[l] = (const float*)d_in[5 + 3 * l];
        gam[l]   = (const float*)d_in[18 + 2 * l];
        bet[l]   = (const float*)d_in[19 + 2 * l];
    }
    const float* Wm[3] = {(const float*)d_in[28], (const float*)d_in[30], (const float*)d_in[32]};
    const float* bm[3] = {(const float*)d_in[29], (const float*)d_in[31], (const float*)d_in[33]};

    // workspace layout
    const size_t bufElems = (size_t)N * 256;
    float* bufA  = (float*)d_ws;
    float* bufB  = bufA + bufElems;
    float* aggb  = bufB + bufElems;
    float* pool  = aggb + bufElems;            // G*128
    float* cnt   = pool + (size_t)G * 128;     // G
    float* m0    = cnt + G;                    // G*128
    float* m1    = m0 + (size_t)G * 128;       // G*64
    float* xpad  = m1 + (size_t)G * 64;        // N*8
    float* w0rp  = xpad + (size_t)N * 8;       // 32*8
    float* w0tp  = w0rp + 256;                 // 32*8
    (void)ws_size; (void)n_in; (void)out_size;

    // ---- pad layer-0 operands (DIN 5 -> 8, zero filled)
    pad5to8_kernel<<<(N * 8 + 255) / 256, 256, 0, stream>>>(x, xpad, N);
    pad5to8_kernel<<<1, 256, 0, stream>>>(Wrel[0], w0rp, 32);
    pad5to8_kernel<<<1, 256, 0, stream>>>(Wroot[0], w0tp, 32);

    const int kshift[5] = {3, 5, 7, 8, 8};     // padded DIN = 8,32,128,256,256
    const float* hs[6];
    hs[0] = xpad;
    float* bufs[2] = {bufA, bufB};

    for (int l = 0; l < 5; ++l) {
        const int din = 1 << kshift[l];
        const int nAgg = N * din;
        zero_kernel<<<(nAgg + 255) / 256, 256, 0, stream>>>(aggb, nAgg);
        const int epb = 256 >> kshift[l];
        scatter_add_kernel<<<(E + epb - 1) / epb, 256, 0, stream>>>(
            hs[l], src, dst, aggb, E, kshift[l]);

        float* outb = bufs[l & 1];
        switch (l) {
            case 0: launch_gcn<8, 32>   (hs[l], aggb, w0rp,    w0tp,     brel[0], gam[0], bet[0], outb, N, stream); break;
            case 1: launch_gcn<32, 128> (hs[l], aggb, Wrel[1], Wroot[1], brel[1], gam[1], bet[1], outb, N, stream); break;
            case 2: launch_gcn<128, 256>(hs[l], aggb, Wrel[2], Wroot[2], brel[2], gam[2], bet[2], outb, N, stream); break;
            case 3: launch_gcn<256, 256>(hs[l], aggb, Wrel[3], Wroot[3], brel[3], gam[3], bet[3], outb, N, stream); break;
            case 4: launch_gcn<256, 128>(hs[l], aggb, Wrel[4], Wroot[4], brel[4], gam[4], bet[4], outb, N, stream); break;
        }
        hs[l + 1] = outb;
    }

    // global mean pool (last hidden dim = 128)
    zero_kernel<<<(G * 128 + 255) / 256, 256, 0, stream>>>(pool, G * 128);
    zero_kernel<<<1, 256, 0, stream>>>(cnt, G);
    pool_accum_kernel<<<N, 128, 0, stream>>>(hs[5], batch, pool, cnt);
    pool_div_kernel<<<(G * 128 + 255) / 256, 256, 0, stream>>>(pool, cnt, G * 128);

    // MLP head: 128 -> 128 -> 64 -> 1
    mlp_kernel<<<G, 128, 0, stream>>>(pool, Wm[0], bm[0], m0, 128, 128, 1);
    mlp_kernel<<<G, 64, 0, stream>>>(m0, Wm[1], bm[1], m1, 128, 64, 1);
    mlp_kernel<<<G, 1, 0, stream>>>(m1, Wm[2], bm[2], (float*)d_out, 64, 1, 0);
}